// MultiHeadGATLayer_17463337025550
// MI455X (gfx1250) — compile-verified
//
#include <hip/hip_runtime.h>
#include <math.h>

typedef __attribute__((ext_vector_type(2))) float v2f;
typedef __attribute__((ext_vector_type(8))) float v8f;

#define IN_DIM 128
#define HID 32
#define HEADS 4
#define PROJ 128   // HEADS*HID
#define OUT_DIM 128
#define TAUF 0.3f

#if __has_builtin(__builtin_amdgcn_wmma_f32_16x16x4_f32)
#define HAS_WMMA_F32 1
#endif

// ---------------------------------------------------------------------------
// Kernel A: fused per-head projections zq, zk, zm  (each [N, 128] row-major)
// One wave computes one 16x16 output tile of one of the three matrices.
// blockDim = 128 (4 waves). blockIdx.x = row tile (N/16), blockIdx.y = 0..5,
// wave picks tile t = blockIdx.y*4 + wave in [0,24): matrix = t/8, ntile = t%8.
// Per-head weight W is [HEADS][IN_DIM][HID]; logical column j = head*32 + e.
// ---------------------------------------------------------------------------
__global__ __launch_bounds__(128) void proj_kernel(
    const float* __restrict__ h,
    const float* __restrict__ Wq, const float* __restrict__ bq,
    const float* __restrict__ Wk, const float* __restrict__ bk,
    const float* __restrict__ Wm, const float* __restrict__ bm,
    float* __restrict__ zq, float* __restrict__ zk, float* __restrict__ zm)
{
    const int lane  = threadIdx.x & 31;
    const int wave  = threadIdx.x >> 5;
    const int mtile = blockIdx.x;
    const int t     = blockIdx.y * 4 + wave;   // 0..23
    const int mat   = t >> 3;
    const int ntile = t & 7;

    const float* W; const float* bias; float* Z;
    if (mat == 0)      { W = Wq; bias = bq; Z = zq; }
    else if (mat == 1) { W = Wk; bias = bk; Z = zk; }
    else               { W = Wm; bias = bm; Z = zm; }

    const int row0 = mtile * 16;
    const int col  = ntile * 16 + (lane & 15);   // output column j in [0,128)
    const int hh   = col >> 5;                   // head
    const int ee   = col & 31;                   // within-head dim
    const int koff = (lane >> 4) << 1;           // 0 (lanes 0-15) or 2 (16-31)

    v8f c = {};
#ifdef HAS_WMMA_F32
    // A(16x4): lane&15 = row; vgpr pair holds K = k0+koff, k0+koff+1
    // B(4x16): lane&15 = col; vgpr pair holds K = k0+koff, k0+koff+1
    const float* __restrict__ Ap = h + (size_t)(row0 + (lane & 15)) * IN_DIM + koff;
    const float* __restrict__ Bp = W + (size_t)hh * (IN_DIM * HID) + (size_t)koff * HID + ee;
    for (int k0 = 0; k0 < IN_DIM; k0 += 4) {
        v2f a, b;
        a.x = Ap[k0];
        a.y = Ap[k0 + 1];
        b.x = Bp[k0 * HID];
        b.y = Bp[(k0 + 1) * HID];
        c = __builtin_amdgcn_wmma_f32_16x16x4_f32(false, a, false, b,
                                                  (short)0, c, false, false);
    }
#else
    // Scalar fallback producing the same C layout
    #pragma unroll
    for (int r = 0; r < 8; ++r) {
        const int row = row0 + ((lane >> 4) << 3) + r;
        float acc = 0.f;
        for (int k = 0; k < IN_DIM; ++k)
            acc += h[(size_t)row * IN_DIM + k] *
                   W[(size_t)hh * (IN_DIM * HID) + (size_t)k * HID + ee];
        c[r] = acc;
    }
#endif
    const float bcol  = bias[col];
    const int   rbase = row0 + ((lane >> 4) << 3);  // lanes>=16 -> rows M+8
    #pragma unroll
    for (int r = 0; r < 8; ++r)
        Z[(size_t)(rbase + r) * PROJ + col] = c[r] + bcol;
}

// ---------------------------------------------------------------------------
// Init: emax (encoded -inf), denom = 0, agg = 0
// ---------------------------------------------------------------------------
__global__ void init_kernel(unsigned int* __restrict__ emax_u,
                            float* __restrict__ denom,
                            float* __restrict__ agg, int nNodes)
{
    const int i  = blockIdx.x * blockDim.x + threadIdx.x;
    const int nh = nNodes * HEADS;
    if (i < nh) { emax_u[i] = 0x007FFFFFu; denom[i] = 0.0f; }  // flip(-inf)
    const int np = nNodes * PROJ;
    if (i < np) agg[i] = 0.0f;
}

__device__ __forceinline__ unsigned int float_flip(float x) {
    unsigned int b = __float_as_uint(x);
    return (b & 0x80000000u) ? ~b : (b | 0x80000000u);
}
__device__ __forceinline__ float float_unflip(unsigned int k) {
    return __uint_as_float((k & 0x80000000u) ? (k & 0x7FFFFFFFu) : ~k);
}

// ---------------------------------------------------------------------------
// Kernel B: e[edge,head] = TAU * dot32(zk[src], zq[dst]); atomicMax into emax
// ---------------------------------------------------------------------------
__global__ __launch_bounds__(256) void edge_logit_kernel(
    const float* __restrict__ zq, const float* __restrict__ zk,
    const int* __restrict__ src, const int* __restrict__ dst,
    float* __restrict__ eArr, unsigned int* __restrict__ emax_u, int nEdges)
{
    const int i = blockIdx.x * blockDim.x + threadIdx.x;
    if (i >= nEdges * HEADS) return;
    const int edge = i >> 2;
    const int hh   = i & 3;
    const int s = src[edge], d = dst[edge];
    const float4* kp = (const float4*)(zk + (size_t)s * PROJ + hh * HID);
    const float4* qp = (const float4*)(zq + (size_t)d * PROJ + hh * HID);
    float acc = 0.f;
    #pragma unroll
    for (int t = 0; t < 8; ++t) {
        float4 a = kp[t], b = qp[t];
        acc += a.x * b.x + a.y * b.y + a.z * b.z + a.w * b.w;
    }
    const float e = TAUF * acc;
    eArr[i] = e;
    atomicMax(&emax_u[(size_t)d * HEADS + hh], float_flip(e));
}

// ---------------------------------------------------------------------------
// Kernel C: eexp = exp(e - emax[dst]); atomicAdd into denom
// ---------------------------------------------------------------------------
__global__ __launch_bounds__(256) void edge_exp_kernel(
    const int* __restrict__ dst, float* __restrict__ eArr,
    const unsigned int* __restrict__ emax_u, float* __restrict__ denom, int nEdges)
{
    const int i = blockIdx.x * blockDim.x + threadIdx.x;
    if (i >= nEdges * HEADS) return;
    const int edge = i >> 2;
    const int hh   = i & 3;
    const int d = dst[edge];
    const float m  = float_unflip(emax_u[(size_t)d * HEADS + hh]);  // finite: d has edges
    const float ex = expf(eArr[i] - m);
    eArr[i] = ex;
    atomicAdd(&denom[(size_t)d * HEADS + hh], ex);
}

// ---------------------------------------------------------------------------
// Kernel C2: alpha = eexp / denom[dst]  (once per edge-head)
// ---------------------------------------------------------------------------
__global__ __launch_bounds__(256) void alpha_kernel(
    const int* __restrict__ dst, float* __restrict__ eArr,
    const float* __restrict__ denom, int nEdges)
{
    const int i = blockIdx.x * blockDim.x + threadIdx.x;
    if (i >= nEdges * HEADS) return;
    const int edge = i >> 2;
    const int hh   = i & 3;
    const float dn = denom[(size_t)dst[edge] * HEADS + hh];
    eArr[i] = eArr[i] / (dn > 0.f ? dn : 1.f);
}

// ---------------------------------------------------------------------------
// Kernel D: agg[dst, j] += alpha[edge, j/32] * zm[src, j]  (fp32 atomics)
// One thread per (edge, component).
// ---------------------------------------------------------------------------
__global__ __launch_bounds__(256) void aggregate_kernel(
    const float* __restrict__ zm, const float* __restrict__ alpha,
    const int* __restrict__ src, const int* __restrict__ dst,
    float* __restrict__ agg, int nEdges)
{
    const int i = blockIdx.x * blockDim.x + threadIdx.x;
    if (i >= nEdges * PROJ) return;               // 204.8M < 2^31
    const int edge = i >> 7;
    const int j    = i & 127;
    const int hh   = j >> 5;
    const int s = src[edge], d = dst[edge];
    const float a = alpha[(size_t)edge * HEADS + hh];
    atomicAdd(&agg[(size_t)d * PROJ + j], a * zm[(size_t)s * PROJ + j]);
}

// ---------------------------------------------------------------------------
// Kernel E: out = relu(agg @ Wout + bout), Wout is [128,128] row-major.
// Same WMMA tiling as kernel A.
// ---------------------------------------------------------------------------
__global__ __launch_bounds__(128) void out_proj_kernel(
    const float* __restrict__ agg, const float* __restrict__ Wout,
    const float* __restrict__ bout, float* __restrict__ out)
{
    const int lane  = threadIdx.x & 31;
    const int wave  = threadIdx.x >> 5;
    const int mtile = blockIdx.x;
    const int ntile = blockIdx.y * 4 + wave;     // 0..7
    const int row0  = mtile * 16;
    const int col   = ntile * 16 + (lane & 15);
    const int koff  = (lane >> 4) << 1;

    v8f c = {};
#ifdef HAS_WMMA_F32
    const float* __restrict__ Ap = agg + (size_t)(row0 + (lane & 15)) * PROJ + koff;
    const float* __restrict__ Bp = Wout + (size_t)koff * OUT_DIM + col;
    for (int k0 = 0; k0 < PROJ; k0 += 4) {
        v2f a, b;
        a.x = Ap[k0];
        a.y = Ap[k0 + 1];
        b.x = Bp[k0 * OUT_DIM];
        b.y = Bp[(k0 + 1) * OUT_DIM];
        c = __builtin_amdgcn_wmma_f32_16x16x4_f32(false, a, false, b,
                                                  (short)0, c, false, false);
    }
#else
    #pragma unroll
    for (int r = 0; r < 8; ++r) {
        const int row = row0 + ((lane >> 4) << 3) + r;
        float acc = 0.f;
        for (int k = 0; k < PROJ; ++k)
            acc += agg[(size_t)row * PROJ + k] * Wout[(size_t)k * OUT_DIM + col];
        c[r] = acc;
    }
#endif
    const float bb    = bout[col];
    const int   rbase = row0 + ((lane >> 4) << 3);
    #pragma unroll
    for (int r = 0; r < 8; ++r)
        out[(size_t)(rbase + r) * OUT_DIM + col] = fmaxf(c[r] + bb, 0.f);
}

// ---------------------------------------------------------------------------
extern "C" void kernel_launch(void* const* d_in, const int* in_sizes, int n_in,
                              void* d_out, int out_size, void* d_ws, size_t ws_size,
                              hipStream_t stream)
{
    const float* h    = (const float*)d_in[0];
    const int*   src  = (const int*)d_in[1];
    const int*   dst  = (const int*)d_in[2];
    const float* Wq   = (const float*)d_in[3];
    const float* bq   = (const float*)d_in[4];
    const float* Wk   = (const float*)d_in[5];
    const float* bk   = (const float*)d_in[6];
    const float* Wm   = (const float*)d_in[7];
    const float* bm   = (const float*)d_in[8];
    const float* Wout = (const float*)d_in[9];
    const float* bout = (const float*)d_in[10];
    float* out = (float*)d_out;

    const int nNodes = in_sizes[0] / IN_DIM;   // 100000
    const int nEdges = in_sizes[1];            // 1600000

    const size_t NP = (size_t)nNodes * PROJ;
    const size_t EH = (size_t)nEdges * HEADS;
    const size_t NH = (size_t)nNodes * HEADS;

    // Workspace layout (fp32 unless noted): zq, zk, zm, agg, e/alpha, emax(u32), denom
    float* zq   = (float*)d_ws;
    float* zk   = zq + NP;
    float* zm   = zk + NP;
    float* agg  = zm + NP;
    float* eArr = agg + NP;
    unsigned int* emax_u = (unsigned int*)(eArr + EH);
    float* denom = (float*)(emax_u + NH);

    // 1. init accumulators
    {
        const int total = (int)NP;   // covers NH too
        init_kernel<<<(total + 255) / 256, 256, 0, stream>>>(emax_u, denom, agg, nNodes);
    }
    // 2. q/k/m projections (WMMA)
    proj_kernel<<<dim3(nNodes / 16, 6), 128, 0, stream>>>(
        h, Wq, bq, Wk, bk, Wm, bm, zq, zk, zm);
    // 3. edge logits + segment max
    const int gEH = (int)((EH + 255) / 256);
    edge_logit_kernel<<<gEH, 256, 0, stream>>>(zq, zk, src, dst, eArr, emax_u, nEdges);
    // 4. exp + segment sum
    edge_exp_kernel<<<gEH, 256, 0, stream>>>(dst, eArr, emax_u, denom, nEdges);
    // 5. normalize to alpha
    alpha_kernel<<<gEH, 256, 0, stream>>>(dst, eArr, denom, nEdges);
    // 6. weighted scatter-sum of messages
    {
        const long long totalAgg = (long long)nEdges * PROJ;
        aggregate_kernel<<<(unsigned)((totalAgg + 255) / 256), 256, 0, stream>>>(
            zm, eArr, src, dst, agg, nEdges);
    }
    // 7. output projection + relu (WMMA)
    out_proj_kernel<<<dim3(nNodes / 16, 2), 128, 0, stream>>>(agg, Wout, bout, out);
}